// GCAM_2680059592817
// MI455X (gfx1250) — compile-verified
//
#include <hip/hip_runtime.h>
#include <math.h>

// GCAM fused pipeline for MI455X (gfx1250, wave32).
// All GEMMs via V_WMMA_F32_16X16X4_F32 (fp32 in/out, memory-bound problem:
// ~56 GFLOP vs ~350MB HBM traffic -> need only ~5 TFLOPS; f32 WMMA suffices
// and preserves exact-precision parity with the fp32 reference).

#define B_   8
#define C_   512
#define CR_  32
#define HW_  4096
#define C2_  1024

typedef __attribute__((ext_vector_type(2))) float v2f;
typedef __attribute__((ext_vector_type(8))) float v8f;

__device__ __forceinline__ v8f wmma4(v2f a, v2f b, v8f c) {
  // D = A(16x4) * B(4x16) + C(16x16), fp32
  return __builtin_amdgcn_wmma_f32_16x16x4_f32(
      /*neg_a=*/false, a, /*neg_b=*/false, b,
      /*c_mod=*/(short)0, c, /*reuse_a=*/false, /*reuse_b=*/false);
}

// ---- wave-level GEMM row: 16 x (NT*16) tile, A row-major [16,K], B row-major [K, N] ----
template<int NT>
__device__ __forceinline__ void gemm_row(const float* __restrict__ A, int lda,
                                         const float* __restrict__ Bm, int ldb,
                                         int K, v8f c[NT]) {
  const int lane = threadIdx.x & 31;
  const int half = lane >> 4;
  const int l16  = lane & 15;
  // A frag: element (m=l16, k=2*half + j) -> contiguous pair
  const float* ap = A + (size_t)l16 * lda + 2 * half;
  // B frag: element (k=2*half + j, n=l16)
  const float* bp = Bm + (size_t)(2 * half) * ldb + l16;
#pragma unroll 2
  for (int k = 0; k < K; k += 4) {
    v2f a = *(const v2f*)ap;
#pragma unroll
    for (int t = 0; t < NT; ++t) {
      v2f b;
      b.x = bp[t * 16];
      b.y = bp[t * 16 + ldb];
      c[t] = wmma4(a, b, c[t]);
    }
    ap += 4;
    bp += (size_t)4 * ldb;
  }
}

// ---- same, but B supplied transposed: computes A[16,K] * Bt[NT*16, K]^T ----
template<int NT>
__device__ __forceinline__ void gemm_row_bt(const float* __restrict__ A, int lda,
                                            const float* __restrict__ Bt, int ldbt,
                                            int K, v8f c[NT]) {
  const int lane = threadIdx.x & 31;
  const int half = lane >> 4;
  const int l16  = lane & 15;
  const float* ap = A + (size_t)l16 * lda + 2 * half;
  const float* bp = Bt + (size_t)l16 * ldbt + 2 * half;
#pragma unroll 2
  for (int k = 0; k < K; k += 4) {
    v2f a = *(const v2f*)ap;
#pragma unroll
    for (int t = 0; t < NT; ++t) {
      v2f b = *(const v2f*)(bp + (size_t)t * 16 * ldbt);
      c[t] = wmma4(a, b, c[t]);
    }
    ap += 4;
    bp += 4;
  }
}

// ================= Kernel 1: h = gelu(W1 . X + b1) =================
__global__ void k_conv1_gelu(const float* __restrict__ x, const float* __restrict__ w1,
                             const float* __restrict__ b1, float* __restrict__ h) {
  constexpr int NT = 4;
  const int wid = blockIdx.x * 8 + (threadIdx.x >> 5);
  const int NG = HW_ / (16 * NT);      // 64
  const int MT = CR_ / 16;             // 2
  const int b  = wid / (MT * NG);
  const int r0 = wid % (MT * NG);
  const int mt = r0 / NG;
  const int ng = r0 % NG;

  const float* A  = w1 + (size_t)mt * 16 * C_;
  const float* Bm = x + (size_t)b * C_ * HW_ + ng * 16 * NT;
  v8f c[NT] = {};
  gemm_row<NT>(A, C_, Bm, HW_, C_, c);

  const int lane = threadIdx.x & 31;
  const int half = lane >> 4;
  const int l16  = lane & 15;
  float* D = h + (size_t)b * CR_ * HW_;
#pragma unroll
  for (int t = 0; t < NT; ++t)
#pragma unroll
    for (int r = 0; r < 8; ++r) {
      const int m = mt * 16 + r + 8 * half;
      const int n = ng * 16 * NT + t * 16 + l16;
      float v = c[t][r] + b1[m];
      v = 0.5f * v * (1.0f + erff(v * 0.70710678118654752f));  // exact erf-GELU
      D[(size_t)m * HW_ + n] = v;
    }
}

// ================= Kernel 2: y = W2 . h + b2 =================
__global__ void k_conv2(const float* __restrict__ h, const float* __restrict__ w2,
                        const float* __restrict__ b2, float* __restrict__ y) {
  constexpr int NT = 4;
  const int wid = blockIdx.x * 8 + (threadIdx.x >> 5);
  const int NG = HW_ / (16 * NT);      // 64
  const int MT = C_ / 16;              // 32
  const int b  = wid / (MT * NG);
  const int r0 = wid % (MT * NG);
  const int mt = r0 / NG;
  const int ng = r0 % NG;

  const float* A  = w2 + (size_t)mt * 16 * CR_;
  const float* Bm = h + (size_t)b * CR_ * HW_ + ng * 16 * NT;
  v8f c[NT] = {};
  gemm_row<NT>(A, CR_, Bm, HW_, CR_, c);

  const int lane = threadIdx.x & 31;
  const int half = lane >> 4;
  const int l16  = lane & 15;
  float* D = y + (size_t)b * C_ * HW_;
#pragma unroll
  for (int t = 0; t < NT; ++t)
#pragma unroll
    for (int r = 0; r < 8; ++r) {
      const int m = mt * 16 + r + 8 * half;
      const int n = ng * 16 * NT + t * 16 + l16;
      D[(size_t)m * HW_ + n] = c[t][r] + b2[m];
    }
}

// ================= Kernel 3: energy = y . y^T =================
__global__ void k_energy(const float* __restrict__ y, float* __restrict__ e) {
  constexpr int NT = 4;
  const int wid = blockIdx.x * 8 + (threadIdx.x >> 5);
  const int NG = C_ / (16 * NT);       // 8
  const int MT = C_ / 16;              // 32
  const int b  = wid / (MT * NG);
  const int r0 = wid % (MT * NG);
  const int mt = r0 / NG;
  const int ng = r0 % NG;

  const float* yb = y + (size_t)b * C_ * HW_;
  const float* A  = yb + (size_t)mt * 16 * HW_;
  const float* Bt = yb + (size_t)ng * 16 * NT * HW_;
  v8f c[NT] = {};
  gemm_row_bt<NT>(A, HW_, Bt, HW_, HW_, c);

  const int lane = threadIdx.x & 31;
  const int half = lane >> 4;
  const int l16  = lane & 15;
  float* D = e + (size_t)b * C_ * C_;
#pragma unroll
  for (int t = 0; t < NT; ++t)
#pragma unroll
    for (int r = 0; r < 8; ++r) {
      const int m = mt * 16 + r + 8 * half;
      const int n = ng * 16 * NT + t * 16 + l16;
      D[(size_t)m * C_ + n] = c[t][r];
    }
}

// ========== Kernel 4: attn = softmax(rowmax - e) == exp(rowmin - e)/sum ==========
__global__ void k_softmax(float* __restrict__ e) {
  __shared__ float red[256];
  const int tid = threadIdx.x;
  float* p = e + (size_t)blockIdx.x * C_;
  const float v0 = p[tid];
  const float v1 = p[tid + 256];
  red[tid] = fminf(v0, v1);
  __syncthreads();
  for (int s = 128; s > 0; s >>= 1) {
    if (tid < s) red[tid] = fminf(red[tid], red[tid + s]);
    __syncthreads();
  }
  const float rmin = red[0];
  __syncthreads();
  const float e0 = expf(rmin - v0);
  const float e1 = expf(rmin - v1);
  red[tid] = e0 + e1;
  __syncthreads();
  for (int s = 128; s > 0; s >>= 1) {
    if (tid < s) red[tid] += red[tid + s];
    __syncthreads();
  }
  const float inv = 1.0f / red[0];
  p[tid] = e0 * inv;
  p[tid + 256] = e1 * inv;
}

// ===== Kernel 5: M2 = gamma * (Wf1 . attn) + Wf1   (exact linear fusion of tail) =====
__global__ void k_m2(const float* __restrict__ attn, const float* __restrict__ wf,
                     const float* __restrict__ gamma, float* __restrict__ m2) {
  constexpr int NT = 4;
  const int wid = blockIdx.x * 8 + (threadIdx.x >> 5);
  const int NG = C_ / (16 * NT);       // 8
  const int MT = C_ / 16;              // 32
  const int b  = wid / (MT * NG);
  const int r0 = wid % (MT * NG);
  const int mt = r0 / NG;
  const int ng = r0 % NG;

  const float* A  = wf + (size_t)mt * 16 * C2_;          // Wf1 rows, lda = 2C
  const float* Bm = attn + (size_t)b * C_ * C_ + ng * 16 * NT;
  v8f c[NT] = {};
  gemm_row<NT>(A, C2_, Bm, C_, C_, c);

  const float g = gamma[0];
  const int lane = threadIdx.x & 31;
  const int half = lane >> 4;
  const int l16  = lane & 15;
  float* D = m2 + (size_t)b * C_ * C_;
#pragma unroll
  for (int t = 0; t < NT; ++t)
#pragma unroll
    for (int r = 0; r < 8; ++r) {
      const int m = mt * 16 + r + 8 * half;
      const int n = ng * 16 * NT + t * 16 + l16;
      D[(size_t)m * C_ + n] = g * c[t][r] + wf[(size_t)m * C2_ + n];
    }
}

// ===== Kernel 6: final = M2 . y + Wf2 . x + bf =====
__global__ void k_fusion(const float* __restrict__ m2, const float* __restrict__ y,
                         const float* __restrict__ x, const float* __restrict__ wf,
                         const float* __restrict__ bf, float* __restrict__ out) {
  constexpr int NT = 4;
  const int wid = blockIdx.x * 8 + (threadIdx.x >> 5);
  const int NG = HW_ / (16 * NT);      // 64
  const int MT = C_ / 16;              // 32
  const int b  = wid / (MT * NG);
  const int r0 = wid % (MT * NG);
  const int mt = r0 / NG;
  const int ng = r0 % NG;

  v8f c[NT] = {};
  // part 1: M2[b] . y[b]
  gemm_row<NT>(m2 + (size_t)b * C_ * C_ + (size_t)mt * 16 * C_, C_,
               y + (size_t)b * C_ * HW_ + ng * 16 * NT, HW_, C_, c);
  // part 2: Wf2 . x[b]  (Wf columns 512..1023)
  gemm_row<NT>(wf + (size_t)mt * 16 * C2_ + C_, C2_,
               x + (size_t)b * C_ * HW_ + ng * 16 * NT, HW_, C_, c);

  const int lane = threadIdx.x & 31;
  const int half = lane >> 4;
  const int l16  = lane & 15;
  float* D = out + (size_t)b * C_ * HW_;
#pragma unroll
  for (int t = 0; t < NT; ++t)
#pragma unroll
    for (int r = 0; r < 8; ++r) {
      const int m = mt * 16 + r + 8 * half;
      const int n = ng * 16 * NT + t * 16 + l16;
      D[(size_t)m * HW_ + n] = c[t][r] + bf[m];
    }
}

extern "C" void kernel_launch(void* const* d_in, const int* in_sizes, int n_in,
                              void* d_out, int out_size, void* d_ws, size_t ws_size,
                              hipStream_t stream) {
  const float* x     = (const float*)d_in[0];
  const float* w1    = (const float*)d_in[1];
  const float* b1    = (const float*)d_in[2];
  const float* w2    = (const float*)d_in[3];
  const float* b2    = (const float*)d_in[4];
  const float* wf    = (const float*)d_in[5];
  const float* bf    = (const float*)d_in[6];
  const float* gamma = (const float*)d_in[7];
  float* out = (float*)d_out;

  char* ws = (char*)d_ws;
  float* h    = (float*)(ws);                                   //  4 MB
  float* y    = (float*)(ws + (size_t)4 * 1024 * 1024);         // 64 MB
  float* attn = (float*)(ws + (size_t)68 * 1024 * 1024);        //  8 MB
  float* m2   = (float*)(ws + (size_t)76 * 1024 * 1024);        //  8 MB  (total 84 MB)

  const dim3 blk(256);  // 8 wave32s
  // waves = B * (M/16) * (N/64); blocks = waves/8
  k_conv1_gelu<<<dim3(128),  blk, 0, stream>>>(x, w1, b1, h);     // 8*2*64   waves
  k_conv2     <<<dim3(2048), blk, 0, stream>>>(h, w2, b2, y);     // 8*32*64  waves
  k_energy    <<<dim3(256),  blk, 0, stream>>>(y, attn);          // 8*32*8   waves
  k_softmax   <<<dim3(4096), blk, 0, stream>>>(attn);             // one row / block
  k_m2        <<<dim3(256),  blk, 0, stream>>>(attn, wf, gamma, m2);
  k_fusion    <<<dim3(2048), blk, 0, stream>>>(m2, y, x, wf, bf, out);
}